// Model_773094114155
// MI455X (gfx1250) — compile-verified
//
#include <hip/hip_runtime.h>
#include <hip/hip_bf16.h>

typedef float v2f __attribute__((ext_vector_type(2)));
typedef float v8f __attribute__((ext_vector_type(8)));

#define CIN  128
#define CHID 128
#define COUT 64

// ---------------- small elementwise kernels ----------------

__global__ void zero_f32(float* __restrict__ p, int n) {
  int i = blockIdx.x * blockDim.x + threadIdx.x;
  if (i < n) p[i] = 0.0f;
}

__global__ void count_deg(const int* __restrict__ dst, float* __restrict__ deg, int E) {
  int e = blockIdx.x * blockDim.x + threadIdx.x;
  if (e < E) atomicAdd(&deg[dst[e]], 1.0f);
}

__global__ void finalize_dinv(float* __restrict__ dinv, int n) {
  int i = blockIdx.x * blockDim.x + threadIdx.x;
  if (i < n) dinv[i] = rsqrtf(dinv[i] + 1.0f);  // self-loop degree
}

// agg[i] = h[i] * dinv[node]^2 + bias[c]   (float4 per thread)
__global__ void self_bias_init(const float* __restrict__ h, const float* __restrict__ dinv,
                               const float* __restrict__ bias, float* __restrict__ agg,
                               int n, int c4shift, int c4mask) {
  int i = blockIdx.x * blockDim.x + threadIdx.x;
  int total = n << c4shift;
  if (i >= total) return;
  int node = i >> c4shift;
  int c4 = (i & c4mask) << 2;
  float di = dinv[node];
  float sc = di * di;
  float4 hv = ((const float4*)h)[i];
  float4 bv = *(const float4*)(bias + c4);
  float4 o;
  o.x = fmaf(hv.x, sc, bv.x);
  o.y = fmaf(hv.y, sc, bv.y);
  o.z = fmaf(hv.z, sc, bv.z);
  o.w = fmaf(hv.w, sc, bv.w);
  ((float4*)agg)[i] = o;
}

__global__ void relu_inplace4(float* __restrict__ p, int n4) {
  int i = blockIdx.x * blockDim.x + threadIdx.x;
  if (i >= n4) return;
  float4 v = ((float4*)p)[i];
  v.x = fmaxf(v.x, 0.0f);
  v.y = fmaxf(v.y, 0.0f);
  v.z = fmaxf(v.z, 0.0f);
  v.w = fmaxf(v.w, 0.0f);
  ((float4*)p)[i] = v;
}

// ---------------- edge scatter-add kernels (L2 float atomics) ----------------
// one wave32 per edge; lanes split the channel dimension

__global__ void edge_agg128(const int* __restrict__ src, const int* __restrict__ dst,
                            const float* __restrict__ dinv, const float* __restrict__ h,
                            float* __restrict__ agg, int E) {
  int t = blockIdx.x * blockDim.x + threadIdx.x;
  int e = t >> 5;
  if (e >= E) return;
  int lane = t & 31;
  int s = src[e], d = dst[e];
  float coef = dinv[s] * dinv[d];
  float4 v = ((const float4*)(h + (size_t)s * CHID))[lane];
  float* o = agg + (size_t)d * CHID + lane * 4;
  atomicAdd(o + 0, v.x * coef);
  atomicAdd(o + 1, v.y * coef);
  atomicAdd(o + 2, v.z * coef);
  atomicAdd(o + 3, v.w * coef);
}

__global__ void edge_agg64(const int* __restrict__ src, const int* __restrict__ dst,
                           const float* __restrict__ dinv, const float* __restrict__ h,
                           float* __restrict__ agg, int E) {
  int t = blockIdx.x * blockDim.x + threadIdx.x;
  int e = t >> 5;
  if (e >= E) return;
  int lane = t & 31;
  int s = src[e], d = dst[e];
  float coef = dinv[s] * dinv[d];
  float2 v = ((const float2*)(h + (size_t)s * COUT))[lane];
  float* o = agg + (size_t)d * COUT + lane * 2;
  atomicAdd(o + 0, v.x * coef);
  atomicAdd(o + 1, v.y * coef);
}

// ---------------- WMMA fp32 GEMMs ----------------
// V_WMMA_F32_16X16X4_F32, D = A(16x4) * B(4x16) + C.
// A frag (2 VGPRs/lane): lane r (r=lane&15), half=lane>>4:
//   a.x = A[r][k0+2*half], a.y = A[r][k0+2*half+1]
// B frag: b.x = B[k0+2*half][col], b.y = B[k0+2*half+1][col], col = n*16 + r
// C/D frag (8 VGPRs): acc[v] = D[m0 + v + 8*half][col]

// h1[100000x128] = x[100000x128] @ W1[128x128]; one wave per 16-row tile, all 8 col-tiles
__global__ __launch_bounds__(256) void gemm1_wmma(const float* __restrict__ x,
                                                  const float* __restrict__ W,
                                                  float* __restrict__ h, int mtiles) {
  __shared__ float wsh[CIN * CHID];  // 64 KB, W1 staged whole
  for (int i = threadIdx.x; i < CIN * CHID; i += 256) wsh[i] = W[i];
  __syncthreads();

  int mtile = blockIdx.x * 8 + (threadIdx.x >> 5);
  if (mtile >= mtiles) return;  // wave-uniform exit: EXEC all-1 for WMMA waves
  const int lane = threadIdx.x & 31;
  const int r = lane & 15;
  const int half = lane >> 4;
  const int m0 = mtile * 16;

  v8f acc[8] = {};
  const float* arow = x + (size_t)(m0 + r) * CIN + 2 * half;
#pragma unroll 4
  for (int k0 = 0; k0 < CIN; k0 += 4) {
    v2f a = *(const v2f*)(arow + k0);
    const float* b0 = &wsh[(k0 + 2 * half) * CHID + r];
#pragma unroll
    for (int n = 0; n < 8; ++n) {
      v2f b;
      b.x = b0[n * 16];
      b.y = b0[n * 16 + CHID];
      acc[n] = __builtin_amdgcn_wmma_f32_16x16x4_f32(false, a, false, b,
                                                     (short)0, acc[n], false, false);
    }
  }
#pragma unroll
  for (int n = 0; n < 8; ++n) {
    float* orow = h + (size_t)(m0 + 8 * half) * CHID + n * 16 + r;
#pragma unroll
    for (int v = 0; v < 8; ++v) orow[(size_t)v * CHID] = acc[n][v];
  }
}

// h2[100000x64] = relu(agg1)[100000x128] @ W2[128x64]; ReLU fused into A loads
__global__ __launch_bounds__(256) void gemm2_wmma(const float* __restrict__ aggIn,
                                                  const float* __restrict__ W,
                                                  float* __restrict__ h2, int mtiles) {
  __shared__ float wsh[CHID * COUT];  // 32 KB
  for (int i = threadIdx.x; i < CHID * COUT; i += 256) wsh[i] = W[i];
  __syncthreads();

  int mtile = blockIdx.x * 8 + (threadIdx.x >> 5);
  if (mtile >= mtiles) return;
  const int lane = threadIdx.x & 31;
  const int r = lane & 15;
  const int half = lane >> 4;
  const int m0 = mtile * 16;

  v8f acc[4] = {};
  const float* arow = aggIn + (size_t)(m0 + r) * CHID + 2 * half;
#pragma unroll 4
  for (int k0 = 0; k0 < CHID; k0 += 4) {
    v2f raw = *(const v2f*)(arow + k0);
    v2f a;
    a.x = fmaxf(raw.x, 0.0f);
    a.y = fmaxf(raw.y, 0.0f);
    const float* b0 = &wsh[(k0 + 2 * half) * COUT + r];
#pragma unroll
    for (int n = 0; n < 4; ++n) {
      v2f b;
      b.x = b0[n * 16];
      b.y = b0[n * 16 + COUT];
      acc[n] = __builtin_amdgcn_wmma_f32_16x16x4_f32(false, a, false, b,
                                                     (short)0, acc[n], false, false);
    }
  }
#pragma unroll
  for (int n = 0; n < 4; ++n) {
    float* orow = h2 + (size_t)(m0 + 8 * half) * COUT + n * 16 + r;
#pragma unroll
    for (int v = 0; v < 8; ++v) orow[(size_t)v * COUT] = acc[n][v];
  }
}

// ---------------- launcher ----------------

extern "C" void kernel_launch(void* const* d_in, const int* in_sizes, int n_in,
                              void* d_out, int out_size, void* d_ws, size_t ws_size,
                              hipStream_t stream) {
  const float* x  = (const float*)d_in[0];
  const int*   ei = (const int*)d_in[1];
  const float* W1 = (const float*)d_in[2];
  const float* b1 = (const float*)d_in[3];
  const float* W2 = (const float*)d_in[4];
  const float* b2 = (const float*)d_in[5];
  float* out = (float*)d_out;

  const int N = in_sizes[0] / CIN;  // 100000
  const int E = in_sizes[1] / 2;    // 3200000
  const int* src = ei;
  const int* dstv = ei + E;

  // workspace layout (floats): dinv | h1 (reused as h2) | agg1   ~103 MB
  float* ws = (float*)d_ws;
  size_t nAlign = ((size_t)N + 255) & ~(size_t)255;
  float* dinv = ws;
  float* h1   = ws + nAlign;
  float* agg1 = h1 + (size_t)N * CHID;
  float* h2   = h1;  // h1 dead after layer-1 edge aggregation

  const int TPB = 256;
  const int mtiles = N / 16;                 // 6250
  const int edgeBlocks = (int)(((long long)E * 32 + TPB - 1) / TPB);

  // degree -> dinv = rsqrt(deg+1)
  zero_f32<<<(N + TPB - 1) / TPB, TPB, 0, stream>>>(dinv, N);
  count_deg<<<(E + TPB - 1) / TPB, TPB, 0, stream>>>(dstv, dinv, E);
  finalize_dinv<<<(N + TPB - 1) / TPB, TPB, 0, stream>>>(dinv, N);

  // layer 1
  gemm1_wmma<<<(mtiles + 7) / 8, TPB, 0, stream>>>(x, W1, h1, mtiles);
  int n4_1 = N * (CHID / 4);
  self_bias_init<<<(n4_1 + TPB - 1) / TPB, TPB, 0, stream>>>(h1, dinv, b1, agg1, N, 5, 31);
  edge_agg128<<<edgeBlocks, TPB, 0, stream>>>(src, dstv, dinv, h1, agg1, E);

  // layer 2 (relu(agg1) fused into gemm2 A-fragment loads)
  gemm2_wmma<<<(mtiles + 7) / 8, TPB, 0, stream>>>(agg1, W2, h2, mtiles);
  int n4_2 = N * (COUT / 4);
  self_bias_init<<<(n4_2 + TPB - 1) / TPB, TPB, 0, stream>>>(h2, dinv, b2, out, N, 4, 15);
  edge_agg64<<<edgeBlocks, TPB, 0, stream>>>(src, dstv, dinv, h2, out, E);
  relu_inplace4<<<(n4_2 + TPB - 1) / TPB, TPB, 0, stream>>>(out, n4_2);
}